// OHEMLoss_2714419331338
// MI455X (gfx1250) — compile-verified
//
#include <hip/hip_runtime.h>
#include <hip/hip_bf16.h>

// ---------------------------------------------------------------------------
// OHEM cross-entropy loss for MI455X (gfx1250, wave32).
// Memory-bound streaming op: one NT pass over 318MB of logits producing a
// 16MB prob-bit array (kept L2-resident), exact 3-stage radix select for the
// rank-k threshold, deterministic double-precision reduction.
// ---------------------------------------------------------------------------

typedef float    f32x4 __attribute__((ext_vector_type(4)));
typedef int      i32x4 __attribute__((ext_vector_type(4)));
typedef unsigned u32x4 __attribute__((ext_vector_type(4)));

constexpr int kC = 19;          // classes (reference fixes C = 19)

// ---------------- init: zero histograms + seed rank k ----------------------
__global__ void ohem_init_kernel(unsigned* __restrict__ ctrl, int nwords,
                                 int kslot, unsigned k0) {
  int i = blockIdx.x * blockDim.x + threadIdx.x;
  if (i < nwords) ctrl[i] = (i == kslot) ? k0 : 0u;
}

// ---------------- tail helper: one pixel, scalar path ----------------------
__device__ __forceinline__ unsigned ohem_one_pixel(const float* __restrict__ logit,
                                                   int i, int HW, int lbl) {
  int b = i / HW;
  int p = i - b * HW;
  const float* base = logit + (size_t)b * (size_t)kC * (size_t)HW + (size_t)p;
  float x[kC];
  float m = -3.402823466e38f;
#pragma unroll
  for (int c = 0; c < kC; ++c) { x[c] = base[(size_t)c * HW]; m = fmaxf(m, x[c]); }
  float s = 0.f;
#pragma unroll
  for (int c = 0; c < kC; ++c) s += expf(x[c] - m);
  float lse = m + logf(s);
  if ((unsigned)lbl >= (unsigned)kC) lbl = 0;   // IGNORE_INDEX (255) -> 0
  float xl = x[0];
#pragma unroll
  for (int c = 1; c < kC; ++c) xl = (c == lbl) ? x[c] : xl;
  return __float_as_uint(expf(xl - lse));
}

// ---------------- pass A: softmax prob + stage-1 histogram (fused) ---------
// 4 pixels per thread: 19 coalesced global_load_b128 (non-temporal so probBits
// survive in the 192MB L2 for the selection passes).
__global__ __launch_bounds__(256)
void ohem_prob_hist_kernel(const float* __restrict__ logit,
                           const int* __restrict__ label,
                           unsigned* __restrict__ probBits,
                           unsigned* __restrict__ hist1,
                           int N, int HW) {
  __shared__ unsigned sh[2048];
  for (int j = threadIdx.x; j < 2048; j += 256) sh[j] = 0u;
  __syncthreads();

  const int i0 = (blockIdx.x * 256 + threadIdx.x) * 4;
  if (i0 + 3 < N) {
    const int b = i0 / HW;
    const int p = i0 - b * HW;   // HW % 4 == 0 so the 4 pixels share b
    const float* base = logit + (size_t)b * (size_t)kC * (size_t)HW + (size_t)p;

    f32x4 v[kC];
#pragma unroll
    for (int c = 0; c < kC; ++c)
      v[c] = __builtin_nontemporal_load((const f32x4*)(base + (size_t)c * HW));

    i32x4 lb = __builtin_nontemporal_load((const i32x4*)(label + i0));

    u32x4 ob;
#pragma unroll
    for (int q = 0; q < 4; ++q) {
      float m = v[0][q];
#pragma unroll
      for (int c = 1; c < kC; ++c) m = fmaxf(m, v[c][q]);
      float s = 0.f;
#pragma unroll
      for (int c = 0; c < kC; ++c) s += expf(v[c][q] - m);
      float lse = m + logf(s);
      int l = lb[q];
      if ((unsigned)l >= (unsigned)kC) l = 0;
      float xl = v[0][q];
#pragma unroll
      for (int c = 1; c < kC; ++c) xl = (c == l) ? v[c][q] : xl;
      unsigned bits = __float_as_uint(expf(xl - lse));
      ob[q] = bits;
      atomicAdd(&sh[bits >> 21], 1u);
    }
    *(u32x4*)(probBits + i0) = ob;      // regular (temporal) store: keep in L2
  } else {
    for (int i = i0; i < N; ++i) {
      unsigned bits = ohem_one_pixel(logit, i, HW, label[i]);
      probBits[i] = bits;
      atomicAdd(&sh[bits >> 21], 1u);
    }
  }

  __syncthreads();
  for (int j = threadIdx.x; j < 2048; j += 256) {
    unsigned c = sh[j];
    if (c) atomicAdd(&hist1[j], c);
  }
}

// ---------------- radix-select stage: pick bin containing rank k -----------
// sel[0] = remaining rank (in/out), sel[1] = bit prefix (in/out),
// sel[2] = final threshold bits (written when finalize != 0).
__global__ void ohem_select_kernel(const unsigned* __restrict__ hist, int nbins,
                                   unsigned* __restrict__ sel, int stageBits,
                                   int finalize, float minThresh) {
  __shared__ unsigned csum[256];
  const int t = threadIdx.x;
  const unsigned k    = sel[0];   // all reads before any write (pre-barrier)
  const unsigned pref = sel[1];
  const int per = nbins >> 8;     // 2048 -> 8, 1024 -> 4
  unsigned mysum = 0;
  for (int j = 0; j < per; ++j) mysum += hist[t * per + j];
  csum[t] = mysum;
  __syncthreads();
  for (int off = 1; off < 256; off <<= 1) {      // Hillis-Steele inclusive scan
    unsigned vv = (t >= off) ? csum[t - off] : 0u;
    __syncthreads();
    csum[t] += vv;
    __syncthreads();
  }
  const unsigned incl = csum[t];
  const unsigned excl = incl - mysum;
  if (excl <= k && k < incl) {                   // exactly one winner thread
    unsigned cum = excl;
    int bsel = t * per;
    for (int j = 0; j < per; ++j) {
      unsigned c = hist[t * per + j];
      if (cum + c > k) { bsel = t * per + j; break; }
      cum += c;
    }
    unsigned prefNew = (pref << stageBits) | (unsigned)bsel;
    sel[0] = k - cum;
    sel[1] = prefNew;
    if (finalize) {
      float thr = __uint_as_float(prefNew);      // exact sorted[k] value
      sel[2] = __float_as_uint(fmaxf(thr, minThresh));
    }
  }
}

// ---------------- refine: histogram of next bit-field under prefix ---------
__global__ __launch_bounds__(256)
void ohem_refine_kernel(const unsigned* __restrict__ probBits, int N,
                        const unsigned* __restrict__ sel,
                        int matchShift, int binShift, unsigned binMask,
                        unsigned* __restrict__ histOut, int nbins) {
  __shared__ unsigned sh[2048];
  for (int j = threadIdx.x; j < nbins; j += 256) sh[j] = 0u;
  __syncthreads();
  const unsigned pref = sel[1];
  const int stride = gridDim.x * 256;
  for (int i = blockIdx.x * 256 + threadIdx.x; i < N; i += stride) {
    __builtin_prefetch(probBits + i + stride, 0, 1);  // global_prefetch_b8
    unsigned b = probBits[i];
    if ((b >> matchShift) == pref)
      atomicAdd(&sh[(b >> binShift) & binMask], 1u);
  }
  __syncthreads();
  for (int j = threadIdx.x; j < nbins; j += 256) {
    unsigned c = sh[j];
    if (c) atomicAdd(&histOut[j], c);
  }
}

// ---------------- reduction: sum(-log(p) * (p < thr)) ----------------------
__global__ __launch_bounds__(256)
void ohem_reduce_kernel(const unsigned* __restrict__ probBits, int N,
                        const unsigned* __restrict__ sel,
                        double* __restrict__ partials) {
  __shared__ double sd[256];
  const float thr = __uint_as_float(sel[2]);
  double acc = 0.0;
  const int stride = gridDim.x * 256;
  for (int i = blockIdx.x * 256 + threadIdx.x; i < N; i += stride) {
    __builtin_prefetch(probBits + i + stride, 0, 1);
    float p = __uint_as_float(probBits[i]);
    if (p < thr) acc += (double)(-logf(fmaxf(p, 1e-37f)));
  }
  sd[threadIdx.x] = acc;
  __syncthreads();
  for (int off = 128; off > 0; off >>= 1) {
    if (threadIdx.x < off) sd[threadIdx.x] += sd[threadIdx.x + off];
    __syncthreads();
  }
  if (threadIdx.x == 0) partials[blockIdx.x] = sd[0];
}

__global__ void ohem_final_kernel(const double* __restrict__ partials, int nparts,
                                  double invDen, float* __restrict__ out) {
  __shared__ double sd[256];
  double acc = 0.0;
  for (int i = threadIdx.x; i < nparts; i += 256) acc += partials[i];
  sd[threadIdx.x] = acc;
  __syncthreads();
  for (int off = 128; off > 0; off >>= 1) {
    if (threadIdx.x < off) sd[threadIdx.x] += sd[threadIdx.x + off];
    __syncthreads();
  }
  if (threadIdx.x == 0) out[0] = (float)(sd[0] * invDen);
}

// ---------------------------------------------------------------------------
extern "C" void kernel_launch(void* const* d_in, const int* in_sizes, int n_in,
                              void* d_out, int out_size, void* d_ws, size_t ws_size,
                              hipStream_t stream) {
  const float* logit = (const float*)d_in[0];   // (B, 19, 512, 1024) f32
  const int*   label = (const int*)d_in[1];     // (B, 512, 1024)

  const int N  = in_sizes[1];                   // B*H*W pixels
  const int HW = 512 * 1024;                    // reference spatial size
  int B = N / HW; if (B < 1) B = 1;
  long long kk = 100000LL * (long long)B;       // MIN_KEPT * B
  if (kk > (long long)N - 1) kk = (long long)N - 1;
  const unsigned k0 = (unsigned)kk;

  // workspace layout (all 4-byte words unless noted)
  unsigned* probBits = (unsigned*)d_ws;         // N words
  unsigned* ctrl     = probBits + N;            // h1|h2|h3|sel = 5128 words
  unsigned* h1  = ctrl;                         // 2048
  unsigned* h2  = h1 + 2048;                    // 2048
  unsigned* h3  = h2 + 2048;                    // 1024
  unsigned* sel = h3 + 1024;                    // 8: [0]=k [1]=pref [2]=thr bits
  double* partials = (double*)(sel + 8);        // 1024 doubles (8B aligned)
  const int RB = 1024;

  ohem_init_kernel<<<(5128 + 255) / 256, 256, 0, stream>>>(ctrl, 5128, 5120, k0);

  const int groups = (N + 3) / 4;
  ohem_prob_hist_kernel<<<(groups + 255) / 256, 256, 0, stream>>>(
      logit, label, probBits, h1, N, HW);

  // stage 1: bits[31:21]
  ohem_select_kernel<<<1, 256, 0, stream>>>(h1, 2048, sel, 11, 0, 0.7f);
  // stage 2: bits[20:10] under prefix
  ohem_refine_kernel<<<RB, 256, 0, stream>>>(probBits, N, sel, 21, 10, 0x7FFu, h2, 2048);
  ohem_select_kernel<<<1, 256, 0, stream>>>(h2, 2048, sel, 11, 0, 0.7f);
  // stage 3: bits[9:0] under prefix -> exact threshold, apply max(thr, 0.7)
  ohem_refine_kernel<<<RB, 256, 0, stream>>>(probBits, N, sel, 10, 0, 0x3FFu, h3, 1024);
  ohem_select_kernel<<<1, 256, 0, stream>>>(h3, 1024, sel, 10, 1, 0.7f);

  ohem_reduce_kernel<<<RB, 256, 0, stream>>>(probBits, N, sel, partials);
  ohem_final_kernel<<<1, 256, 0, stream>>>(partials, RB, 1.0 / (double)N,
                                           (float*)d_out);
}